// SGNet_47330539602646
// MI455X (gfx1250) — compile-verified
//
#include <hip/hip_runtime.h>
#include <hip/hip_bf16.h>
#include <math.h>

typedef __attribute__((ext_vector_type(2))) float v2f;
typedef __attribute__((ext_vector_type(8))) float v8f;

#define F_DIM 128
#define LDS_STRIDE 132   // 132 % 64 == 4 -> 16 distinct banks for column reads

// ---------------------------------------------------------------------------
// deg[i] = 1.0 (self loop)
__global__ __launch_bounds__(256) void k_init_deg(float* __restrict__ deg, int N) {
    int i = blockIdx.x * 256 + threadIdx.x;
    if (i < N) deg[i] = 1.0f;
}

// deg[col[e]] += 1 for real edges
__global__ __launch_bounds__(256) void k_scatter_deg(float* __restrict__ deg,
                                                     const int* __restrict__ ei, int E) {
    int e = blockIdx.x * 256 + threadIdx.x;
    if (e < E) {
        int c = ei[E + e];
        __hip_atomic_fetch_add(&deg[c], 1.0f, __ATOMIC_RELAXED, __HIP_MEMORY_SCOPE_AGENT);
    }
}

// dinv[i] = deg>0 ? rsqrt(deg) : 0  (in place)
__global__ __launch_bounds__(256) void k_finalize_dinv(float* __restrict__ dinv, int N) {
    int i = blockIdx.x * 256 + threadIdx.x;
    if (i < N) {
        float d = dinv[i];
        dinv[i] = (d > 0.0f) ? rsqrtf(d) : 0.0f;
    }
}

// Fold the self-loop term into initialization: h_dst[i,:] = dinv[i]^2 * h_src[i,:]
__global__ __launch_bounds__(256) void k_init_hop(float* __restrict__ hdst,
                                                  const float* __restrict__ hsrc,
                                                  const float* __restrict__ dinv, int N) {
    long long t = (long long)blockIdx.x * 256 + threadIdx.x;
    long long total = (long long)N * F_DIM;
    if (t < total) {
        int node = (int)(t >> 7);
        float di = dinv[node];
        hdst[t] = di * di * hsrc[t];
    }
}

// One wave per edge: h_dst[col,:] += dinv[row]*dinv[col] * h_src[row,:]
// 32 lanes x 4 floats = 128 features. Atomics stay resident in the 192MB L2.
__global__ __launch_bounds__(256) void k_scatter_hop(float* __restrict__ hdst,
                                                     const float* __restrict__ hsrc,
                                                     const float* __restrict__ dinv,
                                                     const int* __restrict__ ei, int E) {
    int e = blockIdx.x * 8 + (threadIdx.x >> 5);
    if (e >= E) return;
    int lane = threadIdx.x & 31;
    int r = ei[e];
    int c = ei[E + e];
    float w = dinv[r] * dinv[c];
    const float4 v = *reinterpret_cast<const float4*>(hsrc + (size_t)r * F_DIM + lane * 4);
    float* dst = hdst + (size_t)c * F_DIM + lane * 4;
    __hip_atomic_fetch_add(dst + 0, w * v.x, __ATOMIC_RELAXED, __HIP_MEMORY_SCOPE_AGENT);
    __hip_atomic_fetch_add(dst + 1, w * v.y, __ATOMIC_RELAXED, __HIP_MEMORY_SCOPE_AGENT);
    __hip_atomic_fetch_add(dst + 2, w * v.z, __ATOMIC_RELAXED, __HIP_MEMORY_SCOPE_AGENT);
    __hip_atomic_fetch_add(dst + 3, w * v.w, __ATOMIC_RELAXED, __HIP_MEMORY_SCOPE_AGENT);
}

// Pre-pack W into exact per-lane WMMA B-operand order:
//   Wp[(((ks*8 + wave)*32 + lane))*2 + j] = W[n*128 + k]
// with  n = wave*16 + (lane&15),  k = ks*4 + 2*(lane>>4) + j.
// Each wave's per-kstep B fetch becomes one contiguous 256B global_load_b64 burst.
__global__ __launch_bounds__(256) void k_pack_w(float* __restrict__ Wp,
                                                const float* __restrict__ W) {
    int t = blockIdx.x * 256 + threadIdx.x;       // 16384 threads
    int j    = t & 1;
    int lane = (t >> 1) & 31;
    int wave = (t >> 6) & 7;
    int ks   = t >> 9;                            // 0..31
    int n = wave * 16 + (lane & 15);
    int k = ks * 4 + 2 * (lane >> 4) + j;
    Wp[t] = W[n * F_DIM + k];
}

// out[m,:] = log_softmax(h[m,:] @ W^T + b)
// Block: 256 threads = 8 waves; block owns rows m0..m0+15 of h AND out
// (same region of d_out: A tile is read into LDS before rows are overwritten).
// Wave w computes the 16x16 tile at columns n0 = 16*w via 32 chained
// V_WMMA_F32_16X16X4_F32 (fp32, full precision).
__global__ __launch_bounds__(256) void k_gemm_logsoftmax(float* __restrict__ out,
                                                         const float* __restrict__ h,
                                                         const float* __restrict__ Wp,
                                                         const float* __restrict__ bias) {
    __shared__ float tile[16 * LDS_STRIDE];

    const int tid = threadIdx.x;
    const int m0 = blockIdx.x * 16;

    // Cooperative, coalesced load of the 16x128 A tile.
    for (int i = tid; i < 16 * F_DIM; i += 256) {
        int r = i >> 7, c = i & 127;
        tile[r * LDS_STRIDE + c] = h[(size_t)(m0 + r) * F_DIM + c];
    }
    __syncthreads();

    const int wave = tid >> 5;       // 0..7 -> output column tile
    const int lane = tid & 31;
    const int n0   = wave * 16;
    const int half = lane >> 4;      // K sub-select (0 -> K+{0,1}, 1 -> K+{2,3})
    const int lm   = lane & 15;      // A row / B column within tile

    v8f acc = {};
    for (int ks = 0; ks < F_DIM / 4; ++ks) {
        v2f a, b;
        // A 16x4: lanes 0-15 hold K=k0+0/k0+1, lanes 16-31 hold K=k0+2/k0+3
        a.x = tile[lm * LDS_STRIDE + ks * 4 + 2 * half + 0];
        a.y = tile[lm * LDS_STRIDE + ks * 4 + 2 * half + 1];
        // B prepacked: contiguous b64 per lane, contiguous 256B per wave
        b = *reinterpret_cast<const v2f*>(Wp + ((((ks * 8 + wave) * 32) + lane) << 1));
        acc = __builtin_amdgcn_wmma_f32_16x16x4_f32(
            /*neg_a=*/false, a, /*neg_b=*/false, b,
            /*c_mod=*/(short)0, acc, /*reuse_a=*/false, /*reuse_b=*/false);
    }
    __syncthreads();  // done reading A tile; reuse LDS for the output tile

    // C/D layout: VGPR v -> row v + 8*(lane/16), col = lane%16
    for (int v = 0; v < 8; ++v) {
        int r = v + 8 * half;
        int col = n0 + lm;
        tile[r * LDS_STRIDE + col] = acc[v] + bias[col];
    }
    __syncthreads();

    // Row-wise log-softmax over 128 values; wave handles rows 2w, 2w+1.
    for (int rr = 0; rr < 2; ++rr) {
        int r = wave * 2 + rr;
        float v0 = tile[r * LDS_STRIDE + lane];
        float v1 = tile[r * LDS_STRIDE + lane + 32];
        float v2 = tile[r * LDS_STRIDE + lane + 64];
        float v3 = tile[r * LDS_STRIDE + lane + 96];
        float mx = fmaxf(fmaxf(v0, v1), fmaxf(v2, v3));
        for (int off = 16; off > 0; off >>= 1)
            mx = fmaxf(mx, __shfl_xor(mx, off, 32));
        float s = expf(v0 - mx) + expf(v1 - mx) + expf(v2 - mx) + expf(v3 - mx);
        for (int off = 16; off > 0; off >>= 1)
            s += __shfl_xor(s, off, 32);
        float lse = mx + logf(s);
        float* orow = out + (size_t)(m0 + r) * F_DIM;
        orow[lane]      = v0 - lse;
        orow[lane + 32] = v1 - lse;
        orow[lane + 64] = v2 - lse;
        orow[lane + 96] = v3 - lse;
    }
}

// ---------------------------------------------------------------------------
extern "C" void kernel_launch(void* const* d_in, const int* in_sizes, int n_in,
                              void* d_out, int out_size, void* d_ws, size_t ws_size,
                              hipStream_t stream) {
    const float* x    = (const float*)d_in[0];   // [N,128]
    const float* W    = (const float*)d_in[1];   // [128,128]
    const float* bias = (const float*)d_in[2];   // [128]
    const int*   ei   = (const int*)d_in[3];     // [2,E]

    const int N = in_sizes[0] / F_DIM;
    const int E = in_sizes[3] / 2;

    // Workspace: dinv[N] | h1[N*128] | Wp[128*128]
    float* dinv = (float*)d_ws;
    float* h1   = dinv + N;                      // 16B-aligned (N*4 % 16 == 0)
    float* Wp   = h1 + (size_t)N * F_DIM;
    float* h2   = (float*)d_out;                 // hop-2 target, then final output

    const int nb_nodes = (N + 255) / 256;
    const int nb_edges = (E + 255) / 256;
    const int nb_feat  = (int)(((long long)N * F_DIM + 255) / 256);
    const int nb_scat  = (E + 7) / 8;            // 1 wave per edge
    const int nb_rows  = N / 16;                 // 100000/16 = 6250 exact

    // gcn_norm
    k_init_deg<<<nb_nodes, 256, 0, stream>>>(dinv, N);
    k_scatter_deg<<<nb_edges, 256, 0, stream>>>(dinv, ei, E);
    k_finalize_dinv<<<nb_nodes, 256, 0, stream>>>(dinv, N);

    // Pack W into per-lane WMMA B-operand order
    k_pack_w<<<(F_DIM * F_DIM) / 256, 256, 0, stream>>>(Wp, W);

    // hop 1: x -> h1
    k_init_hop<<<nb_feat, 256, 0, stream>>>(h1, x, dinv, N);
    k_scatter_hop<<<nb_scat, 256, 0, stream>>>(h1, x, dinv, ei, E);

    // hop 2: h1 -> h2 (stored in d_out)
    k_init_hop<<<nb_feat, 256, 0, stream>>>(h2, h1, dinv, N);
    k_scatter_hop<<<nb_scat, 256, 0, stream>>>(h2, h1, dinv, ei, E);

    // out = log_softmax(h2 @ W^T + b), in place over d_out (block-exclusive rows)
    k_gemm_logsoftmax<<<nb_rows, 256, 0, stream>>>(h2, h2, Wp, bias);
}